// MMConv_74680891343357
// MI455X (gfx1250) — compile-verified
//
#include <hip/hip_runtime.h>
#include <math.h>

typedef __attribute__((ext_vector_type(2))) float v2f;
typedef __attribute__((ext_vector_type(8))) float v8f;

namespace {
constexpr int B_   = 8;
constexpr int CIN_ = 32;
constexpr int COUT_= 32;
constexpr int H_   = 128;   // Wd in reference naming (rows)
constexpr int W_   = 128;   // Hd in reference naming (cols)
constexpr int KK_  = 9;
constexpr int DI_  = 18;    // d_inner
constexpr int HW_  = H_*W_;
constexpr int L_   = H_*W_; // 16384 sequence length
}

__device__ __forceinline__ float siluf(float x)     { return x / (1.f + expf(-x)); }
__device__ __forceinline__ float softplusf(float x) { return x > 20.f ? x : log1pf(expf(x)); }

// sequence index l -> (row along Wd, col along Hd) for two_row_flatten (H even)
__device__ __forceinline__ void l2rc(int l, int& row, int& col) {
  int h2 = l >> 8;            // l / (2*W)
  int r  = l & 1;
  col    = (l >> 1) & 127;
  row    = 2*h2 + r;
}

// ---------------- 1. offset conv 3x3 pad 1 via WMMA fp32 ----------------
// GEMM: M = 16-pixel tile, K = 288 (tap-major: q = tap*32 + ci), N = 18 (co)
__global__ __launch_bounds__(32) void k_offset_conv_wmma(const float* __restrict__ in,
                                                         const float* __restrict__ w,
                                                         const float* __restrict__ bias,
                                                         float* __restrict__ off) {
  int blk  = blockIdx.x;
  int b    = blk >> 10;          // 1024 tiles per batch
  int tile = blk & 1023;
  int pbase = tile*16;
  __shared__ float lds[16][292];
  int lane = threadIdx.x;

  // phase 1: im2col 16 pixels x 9 taps x 32 input channels (zero pad at borders)
  for (int idx = lane; idx < 16*KK_; idx += 32) {
    int pix = idx / KK_, tap = idx % KK_;
    int p = pbase + pix;
    int y = p >> 7, x = p & 127;
    int dy = tap / 3, dx = tap % 3;
    int iy = y + dy - 1, ix = x + dx - 1;
    bool ok = ((unsigned)iy < (unsigned)H_) && ((unsigned)ix < (unsigned)W_);
    for (int ci = 0; ci < CIN_; ++ci) {
      float v = 0.f;
      if (ok) v = in[((size_t)(b*CIN_ + ci))*HW_ + iy*W_ + ix];
      lds[pix][tap*32 + ci] = v;
    }
  }
  __syncthreads();

  // phase 2: 16 x 32(co, 18 valid) x 288 with V_WMMA_F32_16X16X4_F32
  int m  = lane & 15;
  int hi = lane >> 4;
  v8f c0 = {0.f,0.f,0.f,0.f,0.f,0.f,0.f,0.f};
  v8f c1 = {0.f,0.f,0.f,0.f,0.f,0.f,0.f,0.f};
  for (int q0 = 0; q0 < 288; q0 += 4) {
    int qa = q0 + (hi ? 2 : 0);
    int qb = q0 + (hi ? 3 : 1);
    v2f a;
    a.x = lds[m][qa];
    a.y = lds[m][qb];
    int cia = qa & 31, ka = qa >> 5;   // ka = tap index
    int cib = qb & 31, kb = qb >> 5;
    v2f b0, b1;
    b0.x = w[(size_t)m*288 + cia*9 + ka];
    b0.y = w[(size_t)m*288 + cib*9 + kb];
    b1.x = (m < 2) ? w[(size_t)(m+16)*288 + cia*9 + ka] : 0.f;   // co >= 18 -> zero
    b1.y = (m < 2) ? w[(size_t)(m+16)*288 + cib*9 + kb] : 0.f;
    c0 = __builtin_amdgcn_wmma_f32_16x16x4_f32(false, a, false, b0, (short)0, c0, false, false);
    c1 = __builtin_amdgcn_wmma_f32_16x16x4_f32(false, a, false, b1, (short)0, c1, false, false);
  }

  // phase 3: scatter (only co < 18 stored)
#pragma unroll
  for (int r = 0; r < 8; ++r) {
    int pix = r + (hi ? 8 : 0);
    int p = pbase + pix;
    int y = p >> 7, x = p & 127;
    off[((size_t)(b*2*KK_ + m))*HW_ + y*W_ + x] = c0[r] + bias[m];
    if (m < 2)
      off[((size_t)(b*2*KK_ + 16 + m))*HW_ + y*W_ + x] = c1[r] + bias[16 + m];
  }
}

// ---------------- generic GroupNorm stats ----------------
__global__ void k_gn_stats(const float* __restrict__ x, float* __restrict__ mean,
                           float* __restrict__ rstd, int cpg, int Ctot) {
  int bg = blockIdx.x;
  int G  = Ctot / cpg;
  int b  = bg / G, g = bg % G;
  const float* base = x + ((size_t)(b*Ctot + g*cpg))*HW_;
  int N = cpg*HW_;
  float s = 0.f, ss = 0.f;
  for (int i = threadIdx.x; i < N; i += blockDim.x) { float v = base[i]; s += v; ss += v*v; }
  __shared__ float sh1[256], sh2[256];
  sh1[threadIdx.x] = s; sh2[threadIdx.x] = ss; __syncthreads();
  for (int o = blockDim.x >> 1; o > 0; o >>= 1) {
    if (threadIdx.x < o) { sh1[threadIdx.x] += sh1[threadIdx.x+o]; sh2[threadIdx.x] += sh2[threadIdx.x+o]; }
    __syncthreads();
  }
  if (threadIdx.x == 0) {
    float mu  = sh1[0] / N;
    float var = sh2[0] / N - mu*mu;
    mean[bg] = mu;
    rstd[bg] = rsqrtf(var + 1e-5f);
  }
}

// ---------------- 2. GN + tanh on offset (in place) ----------------
__global__ void k_gn_tanh(float* __restrict__ off, const float* __restrict__ mean,
                          const float* __restrict__ rstd, const float* __restrict__ gw,
                          const float* __restrict__ gb) {
  int idx = blockIdx.x*blockDim.x + threadIdx.x;
  const int total = B_*2*KK_*HW_;
  if (idx >= total) return;
  int c  = (idx / HW_) % (2*KK_);
  int b  = idx / (2*KK_*HW_);
  int bg = b*KK_ + (c >> 1);
  float v = off[idx];
  off[idx] = tanhf((v - mean[bg]) * rstd[bg] * gw[c] + gb[c]);
}

// ---------------- 3. xz = token @ in_proj^T ----------------
__global__ void k_xz(const float* __restrict__ off, const float* __restrict__ ipw,
                     float* __restrict__ xz) {
  int idx = blockIdx.x*blockDim.x + threadIdx.x;
  const int total = B_*L_*36;
  if (idx >= total) return;
  int j = idx % 36; int t = idx / 36;
  int l = t % L_;   int b = t / L_;
  int row, col; l2rc(l, row, col);
  float acc = 0.f;
#pragma unroll
  for (int c = 0; c < 9; ++c)
    acc += off[((size_t)(b*2*KK_ + c))*HW_ + row*W_ + col] * ipw[j*9 + c];
  xz[idx] = acc;
}

// ---------------- 4. causal dwconv1d + silu per direction ----------------
__global__ void k_conv1d(const float* __restrict__ xz, const float* __restrict__ cw,
                         const float* __restrict__ cb, float* __restrict__ xi) {
  int idx = blockIdx.x*blockDim.x + threadIdx.x;
  const int total = 2*B_*L_*DI_;
  if (idx >= total) return;
  int c = idx % DI_; int t = idx / DI_;
  int l = t % L_;    t /= L_;
  int b = t % B_;    int dir = t / B_;
  float acc = cb[c];
#pragma unroll
  for (int tt = 0; tt < 4; ++tt) {
    int j = l - 3 + tt;
    if (j >= 0) {
      int ol = dir ? (L_-1-j) : j;
      acc += cw[c*4+tt] * xz[((size_t)b*L_ + ol)*36 + c];
    }
  }
  xi[idx] = siluf(acc);
}

// ---------------- 5. dt / B / C from x_dbl ----------------
__global__ void k_xdbl(const float* __restrict__ xi, const float* __restrict__ xpw,
                       const float* __restrict__ dtw, const float* __restrict__ dtbias,
                       float* __restrict__ dtb, float* __restrict__ Bb, float* __restrict__ Cb) {
  int idx = blockIdx.x*blockDim.x + threadIdx.x;
  const int total = 2*B_*L_*50;
  if (idx >= total) return;
  int j = idx % 50; int t = idx / 50;
  int l = t % L_;   t /= L_;
  int b = t % B_;   int dir = t / B_;
  size_t tok = ((size_t)dir*B_ + b)*L_ + l;
  const float* xv = xi + tok*DI_;
  if (j < 18) {
    float x0 = 0.f;
#pragma unroll
    for (int d = 0; d < DI_; ++d) x0 += xv[d] * xpw[d];           // row 0
    dtb[tok*DI_ + j] = softplusf(x0 * dtw[j] + dtbias[j]);
  } else if (j < 34) {
    int n = j - 18; float a = 0.f;
#pragma unroll
    for (int d = 0; d < DI_; ++d) a += xv[d] * xpw[(1+n)*DI_ + d];
    Bb[tok*16 + n] = a;
  } else {
    int n = j - 34; float a = 0.f;
#pragma unroll
    for (int d = 0; d < DI_; ++d) a += xv[d] * xpw[(17+n)*DI_ + d];
    Cb[tok*16 + n] = a;
  }
}

// ---------------- 6. selective scan (sequential over L) ----------------
__global__ __launch_bounds__(288) void k_scan(const float* __restrict__ xi,
                                              const float* __restrict__ dtb,
                                              const float* __restrict__ Bb,
                                              const float* __restrict__ Cb,
                                              const float* __restrict__ xz,
                                              const float* __restrict__ A_log,
                                              const float* __restrict__ A_b_log,
                                              const float* __restrict__ Dp,
                                              float* __restrict__ yb) {
  int blk = blockIdx.x;
  int dir = blk / B_, b = blk % B_;
  int tid = threadIdx.x;
  int d = tid >> 4, n = tid & 15;
  const float* al = dir ? A_b_log : A_log;
  float Aa = -expf(al[d*16 + n]);
  float Dd = Dp[d];
  __shared__ float su[18], sdt[18], sB[16], sC[16], sz[18];
  float h = 0.f;
  size_t base = ((size_t)dir*B_ + b)*L_;
  for (int l = 0; l < L_; ++l) {
    size_t t18 = (base + l)*DI_;
    size_t t16 = (base + l)*16;
    int ol = dir ? (L_-1-l) : l;
    // prefetch 8 iterations ahead (sequential streams; hides HBM latency
    // inside the serial recurrence) -> global_prefetch_b8
    if (l + 8 < L_) {
      if (tid == 0)      __builtin_prefetch(xi  + (base + l + 8)*DI_, 0, 1);
      else if (tid == 1) __builtin_prefetch(dtb + (base + l + 8)*DI_, 0, 1);
      else if (tid == 2) __builtin_prefetch(Bb  + (base + l + 8)*16, 0, 1);
      else if (tid == 3) __builtin_prefetch(Cb  + (base + l + 8)*16, 0, 1);
    }
    if (tid < 18)       su[tid]     = xi [t18 + tid];
    else if (tid < 36)  sdt[tid-18] = dtb[t18 + tid - 18];
    else if (tid < 52)  sB [tid-36] = Bb [t16 + tid - 36];
    else if (tid < 68)  sC [tid-52] = Cb [t16 + tid - 52];
    else if (tid < 86)  sz [tid-68] = xz [((size_t)b*L_ + ol)*36 + 18 + (tid - 68)];
    __syncthreads();
    float dtv = sdt[d], u = su[d];
    float dA = __expf(dtv * Aa);
    h = dA*h + dtv * sB[n] * u;
    float v = h * sC[n];
    for (int o = 8; o > 0; o >>= 1) v += __shfl_down(v, o, 16);
    if (n == 0) {
      float y = v + u*Dd;
      float z = sz[d];
      yb[(((size_t)dir*B_ + b)*L_ + ol)*DI_ + d] = y * siluf(z);
    }
    __syncthreads();
  }
}

// ---------------- 7. combine dirs, out_proj, inv flatten, y = w*yk + y_new ----------------
__global__ void k_combine(const float* __restrict__ yb, const float* __restrict__ opw,
                          const float* __restrict__ off, const float* __restrict__ altho,
                          float* __restrict__ ymap) {
  int idx = blockIdx.x*blockDim.x + threadIdx.x;
  const int total = B_*L_*KK_;
  if (idx >= total) return;
  int c = idx % KK_; int t = idx / KK_;
  int l = t % L_;    int b = t / L_;
  const float* y0 = yb + ((size_t)b*L_ + l)*DI_;
  const float* y1 = yb + (((size_t)B_ + b)*L_ + l)*DI_;
  float s = 0.f;
#pragma unroll
  for (int d = 0; d < DI_; ++d) s += (y0[d] + y1[d]) * opw[c*DI_ + d];
  int row, col; l2rc(l, row, col);
  float cum = 0.f;
  if (c < 4)      { for (int j = c; j <= 3; ++j) cum += off[((size_t)(b*2*KK_+j))*HW_ + row*W_ + col]; }
  else if (c > 4) { for (int j = 5; j <= c; ++j) cum += off[((size_t)(b*2*KK_+j))*HW_ + row*W_ + col]; }
  float ynew = (float)row + cum;                       // EXTEND_SCOPE = 1
  float wv = fmaxf(softplusf(altho[0]), 0.01f);
  ymap[((size_t)(b*KK_ + c))*HW_ + row*W_ + col] = wv*s + ynew;
}

// ---------------- 8. fused deformable sample + (9,1) stride-9 conv via WMMA fp32 ----------------
__global__ __launch_bounds__(32) void k_deform_wmma(const float* __restrict__ in,
                                                    const float* __restrict__ ymap,
                                                    const float* __restrict__ dscw,
                                                    const float* __restrict__ dscb,
                                                    float* __restrict__ pre) {
  int blk  = blockIdx.x;
  int b    = blk >> 10;          // 1024 tiles per batch
  int tile = blk & 1023;
  int pbase = tile*16;
  __shared__ float lds[16][292];
  int lane = threadIdx.x;

  // phase 1: bilinear gather of 16 pixels x 9 taps x 32 input channels
  for (int idx = lane; idx < 16*KK_; idx += 32) {
    int pix = idx / KK_, tap = idx % KK_;
    int p = pbase + pix;
    int wrow = p >> 7, hd = p & 127;
    float gx = fminf(fmaxf((float)hd + (float)(tap - 4), 0.f), 127.f);
    float gy = fminf(fmaxf(ymap[((size_t)(b*KK_ + tap))*HW_ + wrow*W_ + hd], 0.f), 127.f);
    float x0f = floorf(gx), y0f = floorf(gy);
    int ix0 = (int)x0f, iy0 = (int)y0f;
    float fx = gx - x0f, fy = gy - y0f;
    float vx1 = (ix0 + 1 <= 127) ? 1.f : 0.f;
    float vy1 = (iy0 + 1 <= 127) ? 1.f : 0.f;
    int ix1 = min(ix0 + 1, 127), iy1 = min(iy0 + 1, 127);
    float w00 = (1.f-fx)*(1.f-fy);
    float w01 = fx*vx1*(1.f-fy);
    float w10 = (1.f-fx)*fy*vy1;
    float w11 = fx*vx1*fy*vy1;
    for (int ci = 0; ci < CIN_; ++ci) {
      const float* ib = in + ((size_t)(b*CIN_ + ci))*HW_;
      float sm = w00*ib[iy0*W_+ix0] + w01*ib[iy0*W_+ix1]
               + w10*ib[iy1*W_+ix0] + w11*ib[iy1*W_+ix1];
      lds[pix][tap*32 + ci] = sm;                // K index q = tap*32 + ci
    }
  }
  __syncthreads();

  // phase 2: GEMM 16(pixels) x 32(co) x 288 via V_WMMA_F32_16X16X4_F32
  int m  = lane & 15;
  int hi = lane >> 4;
  v8f c0 = {0.f,0.f,0.f,0.f,0.f,0.f,0.f,0.f};
  v8f c1 = {0.f,0.f,0.f,0.f,0.f,0.f,0.f,0.f};
  for (int q0 = 0; q0 < 288; q0 += 4) {
    int qa = q0 + (hi ? 2 : 0);
    int qb = q0 + (hi ? 3 : 1);
    v2f a;
    a.x = lds[m][qa];
    a.y = lds[m][qb];
    int cia = qa & 31, ka = qa >> 5;
    int cib = qb & 31, kb = qb >> 5;
    v2f b0, b1;
    b0.x = dscw[(size_t)m*288       + cia*9 + ka];
    b0.y = dscw[(size_t)m*288       + cib*9 + kb];
    b1.x = dscw[(size_t)(m+16)*288  + cia*9 + ka];
    b1.y = dscw[(size_t)(m+16)*288  + cib*9 + kb];
    c0 = __builtin_amdgcn_wmma_f32_16x16x4_f32(false, a, false, b0, (short)0, c0, false, false);
    c1 = __builtin_amdgcn_wmma_f32_16x16x4_f32(false, a, false, b1, (short)0, c1, false, false);
  }

  // phase 3: scatter D (M=pixel, N=co) + bias
#pragma unroll
  for (int r = 0; r < 8; ++r) {
    int pix = r + (hi ? 8 : 0);
    int p = pbase + pix;
    int wrow = p >> 7, hd = p & 127;
    pre[((size_t)(b*COUT_ + m     ))*HW_ + wrow*W_ + hd] = c0[r] + dscb[m];
    pre[((size_t)(b*COUT_ + m + 16))*HW_ + wrow*W_ + hd] = c1[r] + dscb[m+16];
  }
}

// ---------------- 9. final GroupNorm apply ----------------
__global__ void k_gn2_apply(const float* __restrict__ pre, const float* __restrict__ mean,
                            const float* __restrict__ rstd, const float* __restrict__ gw,
                            const float* __restrict__ gb, float* __restrict__ out) {
  int idx = blockIdx.x*blockDim.x + threadIdx.x;
  const int total = B_*COUT_*HW_;
  if (idx >= total) return;
  int c  = (idx / HW_) % COUT_;
  int b  = idx / (COUT_*HW_);
  int bg = b*8 + (c >> 2);
  out[idx] = (pre[idx] - mean[bg]) * rstd[bg] * gw[c] + gb[c];
}

extern "C" void kernel_launch(void* const* d_in, const int* in_sizes, int n_in,
                              void* d_out, int out_size, void* d_ws, size_t ws_size,
                              hipStream_t stream) {
  const float* input   = (const float*)d_in[0];
  const float* ocw     = (const float*)d_in[1];
  const float* ocb     = (const float*)d_in[2];
  const float* gow     = (const float*)d_in[3];
  const float* gob     = (const float*)d_in[4];
  const float* ipw     = (const float*)d_in[5];
  const float* c1w     = (const float*)d_in[6];
  const float* c1b     = (const float*)d_in[7];
  const float* xpw     = (const float*)d_in[8];
  const float* dtw     = (const float*)d_in[9];
  const float* dtbias  = (const float*)d_in[10];
  const float* A_log   = (const float*)d_in[11];
  const float* A_b_log = (const float*)d_in[12];
  const float* D_ssm   = (const float*)d_in[13];
  const float* opw     = (const float*)d_in[14];
  const float* altho   = (const float*)d_in[15];
  const float* dscw    = (const float*)d_in[16];
  const float* dscb    = (const float*)d_in[17];
  const float* gnw     = (const float*)d_in[18];
  const float* gnb     = (const float*)d_in[19];
  float* out = (float*)d_out;

  // workspace layout (floats)
  float* ws = (float*)d_ws;
  size_t o = 0;
  float* off  = ws + o; o += (size_t)B_*2*KK_*HW_;          // 2,359,296
  float* m1   = ws + o; o += 128;
  float* r1   = ws + o; o += 128;
  float* xz   = ws + o; o += (size_t)B_*L_*36;              // 4,718,592
  float* xi   = ws + o; o += (size_t)2*B_*L_*DI_;           // 4,718,592
  float* dtb  = ws + o; o += (size_t)2*B_*L_*DI_;           // 4,718,592
  float* Bb   = ws + o; o += (size_t)2*B_*L_*16;            // 4,194,304
  float* Cb   = ws + o; o += (size_t)2*B_*L_*16;            // 4,194,304
  float* yb   = ws + o; o += (size_t)2*B_*L_*DI_;           // 4,718,592
  float* ymap = ws + o; o += (size_t)B_*KK_*HW_;            // 1,179,648
  float* pre  = ws + o; o += (size_t)B_*COUT_*HW_;          // 4,194,304
  float* m2   = ws + o; o += 128;
  float* r2   = ws + o; o += 128;
  if (ws_size < o * sizeof(float)) return;

  const int TB = 256;
  auto cdiv = [](int a, int b){ return (a + b - 1) / b; };

  // 1. offset conv (WMMA fp32, im2col in LDS)
  k_offset_conv_wmma<<<B_*1024, 32, 0, stream>>>(input, ocw, ocb, off);
  // 2. GN(K groups) + tanh
  k_gn_stats<<<B_*KK_, TB, 0, stream>>>(off, m1, r1, 2, 2*KK_);
  k_gn_tanh<<<cdiv(B_*2*KK_*HW_, TB), TB, 0, stream>>>(off, m1, r1, gow, gob);
  // 3. in_proj
  k_xz<<<cdiv(B_*L_*36, TB), TB, 0, stream>>>(off, ipw, xz);
  // 4. causal dwconv1d + silu, both directions
  k_conv1d<<<cdiv(2*B_*L_*DI_, TB), TB, 0, stream>>>(xz, c1w, c1b, xi);
  // 5. dt / B / C
  k_xdbl<<<cdiv(2*B_*L_*50, TB), TB, 0, stream>>>(xi, xpw, dtw, dtbias, dtb, Bb, Cb);
  // 6. selective scan (one block per (dir, batch))
  k_scan<<<2*B_, 288, 0, stream>>>(xi, dtb, Bb, Cb, xz, A_log, A_b_log, D_ssm, yb);
  // 7. combine + out_proj + y_new
  k_combine<<<cdiv(B_*L_*KK_, TB), TB, 0, stream>>>(yb, opw, off, altho, ymap);
  // 8. fused deformable sample + stride-9 conv (WMMA fp32)
  k_deform_wmma<<<B_*1024, 32, 0, stream>>>(input, ymap, dscw, dscb, pre);
  // 9. final GN
  k_gn_stats<<<B_*8, TB, 0, stream>>>(pre, m2, r2, 4, COUT_);
  k_gn2_apply<<<cdiv(B_*COUT_*HW_, TB), TB, 0, stream>>>(pre, m2, r2, gnw, gnb, out);

  (void)in_sizes; (void)n_in; (void)out_size;
}